// VGCN_RetDistros_64862596104306
// MI455X (gfx1250) — compile-verified
//
#include <hip/hip_runtime.h>
#include <hip/hip_bf16.h>

typedef float v2f __attribute__((ext_vector_type(2)));
typedef float v8f __attribute__((ext_vector_type(8)));

#define N_NODES 100000

// ---------------------------------------------------------------------------
// Elementwise / degree kernels
// ---------------------------------------------------------------------------
__global__ void fill_ones(float* p, long long n) {
    long long i = (long long)blockIdx.x * blockDim.x + threadIdx.x;
    if (i < n) p[i] = 1.0f;
}

__global__ void edge_degree(const long long* __restrict__ dst,
                            const float* __restrict__ ew,
                            float* __restrict__ deg1, float* __restrict__ deg2,
                            long long E) {
    long long e = (long long)blockIdx.x * blockDim.x + threadIdx.x;
    if (e >= E) return;
    long long d = dst[e];
    unsafeAtomicAdd(&deg1[d], ew[e]);   // global_atomic_add_f32 (no return)
    unsafeAtomicAdd(&deg2[d], 1.0f);
}

__global__ void rsqrt_inplace(float* p, long long n) {
    long long i = (long long)blockIdx.x * blockDim.x + threadIdx.x;
    if (i >= n) return;
    float v = p[i];
    p[i] = (v > 0.0f) ? rsqrtf(v) : 0.0f;
}

template <int F>
__global__ void bias_relu(float* __restrict__ h, const float* __restrict__ b,
                          long long total) {
    long long i = (long long)blockIdx.x * blockDim.x + threadIdx.x;
    if (i >= total) return;
    h[i] = fmaxf(h[i] + b[i & (F - 1)], 0.0f);
}

template <int F>
__global__ void bias_tail(float* __restrict__ out, const float* __restrict__ b,
                          long long total) {
    long long i = (long long)blockIdx.x * blockDim.x + threadIdx.x;
    if (i < total) out[i] += b[i & (F - 1)];
    if (i == 0) { out[total] = 0.0f; out[total + 1] = 0.0f; }
}

// ---------------------------------------------------------------------------
// Edge scatter: out[dst] += t[src] * (dinv[src] * w * dinv[dst])
// 4 features per thread: b128 gather + 4 global_atomic_add_f32 (L2-resident).
// ---------------------------------------------------------------------------
template <int F>
__global__ void scatter_edges(const float* __restrict__ t,
                              const long long* __restrict__ src,
                              const long long* __restrict__ dst,
                              const float* __restrict__ ew,   // nullptr => 1.0
                              const float* __restrict__ dinv,
                              float* __restrict__ out, long long E) {
    constexpr int CH = F / 4;  // chunks of 4 floats per edge
    long long idx = (long long)blockIdx.x * blockDim.x + threadIdx.x;
    if (idx >= E * CH) return;
    long long e = idx / CH;
    int f = (int)(idx % CH) * 4;
    long long s = src[e];
    long long d = dst[e];
    float w = ew ? ew[e] : 1.0f;
    float norm = dinv[s] * w * dinv[d];
    const float4 v = *(const float4*)(t + s * F + f);
    float* o = out + d * F + f;
    unsafeAtomicAdd(o + 0, v.x * norm);
    unsafeAtomicAdd(o + 1, v.y * norm);
    unsafeAtomicAdd(o + 2, v.z * norm);
    unsafeAtomicAdd(o + 3, v.w * norm);
}

// ---------------------------------------------------------------------------
// FP32 WMMA GEMM with fused self-loop epilogue:
//   T[M x Fout]   = A[M x 128] * B[128 x Fout]          (kept for edge scatter)
//   Acc[M x Fout] = T * dinv[row]^2                     (accumulator init)
// One wave: 16-row stripe x all NT column tiles, K-loop step 4.
// V_WMMA_F32_16X16X4_F32 VGPR layout (ISA 7.12.2):
//   A 16x4 : lanes 0-15 -> {K=k, K=k+1}, lanes 16-31 -> {K=k+2, K=k+3}, M=lane&15
//   B 4x16 : mirrored (K in VGPR/lane-half, N=lane&15)
//   C 16x16: VGPR r -> row r (lanes 0-15) / row r+8 (lanes 16-31), N=lane&15
// ---------------------------------------------------------------------------
template <int NT>
__global__ void gemm_wmma_f32(const float* __restrict__ A,
                              const float* __restrict__ B,
                              const float* __restrict__ dinv,
                              float* __restrict__ T,
                              float* __restrict__ Acc, int M) {
    constexpr int K = 128;
    constexpr int Fout = NT * 16;
    const int wave = threadIdx.x >> 5;
    const int lane = threadIdx.x & 31;
    const int rowTile = blockIdx.x * (blockDim.x >> 5) + wave;
    if (rowTile * 16 >= M) return;          // wave-uniform: EXEC stays all-ones
    const int row0 = rowTile * 16;
    const int lm = lane & 15;               // M (for A) / N (for B,C) within tile
    const int lh = lane >> 4;               // lane half selects K pair / M+8

    v8f acc[NT];
#pragma unroll
    for (int t = 0; t < NT; ++t) acc[t] = {};

    for (int k = 0; k < K; k += 4) {
        const float* ap = A + (size_t)(row0 + lm) * K + k + 2 * lh;
        v2f a;
        a.x = ap[0];
        a.y = ap[1];
#pragma unroll
        for (int t = 0; t < NT; ++t) {
            const float* bp = B + (size_t)(k + 2 * lh) * Fout + t * 16 + lm;
            v2f b;
            b.x = bp[0];
            b.y = bp[Fout];
            acc[t] = __builtin_amdgcn_wmma_f32_16x16x4_f32(
                false, a, false, b, (short)0, acc[t], false, false);
        }
    }

    // Epilogue: store raw product + dinv^2-scaled accumulator seed.
#pragma unroll
    for (int r = 0; r < 8; ++r) {
        const int row = row0 + 8 * lh + r;
        const float di = dinv[row];
        const float sc = di * di;
#pragma unroll
        for (int t = 0; t < NT; ++t) {
            const float val = acc[t][r];
            const size_t off = (size_t)row * Fout + t * 16 + lm;
            T[off] = val;
            Acc[off] = val * sc;
        }
    }
}

// ---------------------------------------------------------------------------
// Host-side orchestration
// ---------------------------------------------------------------------------
static inline unsigned int gridFor(long long work, int block) {
    return (unsigned int)((work + block - 1) / block);
}

extern "C" void kernel_launch(void* const* d_in, const int* in_sizes, int n_in,
                              void* d_out, int out_size, void* d_ws, size_t ws_size,
                              hipStream_t stream) {
    const float*     x  = (const float*)d_in[0];
    const long long* ei = (const long long*)d_in[1];   // int64 [2, E]
    const float*     ew = (const float*)d_in[2];
    const float*     W1 = (const float*)d_in[3];
    const float*     b1 = (const float*)d_in[4];
    const float*     W2 = (const float*)d_in[5];
    const float*     b2 = (const float*)d_in[6];

    const long long N = N_NODES;
    const long long E = in_sizes[2];
    const long long* src = ei;
    const long long* dst = ei + E;

    float* ws    = (float*)d_ws;
    float* dinv1 = ws;                       // N   (deg1 then dinv1)
    float* dinv2 = ws + N;                   // N   (deg2 then dinv2)
    float* t1    = ws + 2 * N;               // N*128  (x@W1, reused as t2)
    float* h     = t1 + N * 128;             // N*128  (layer-1 accumulator)
    float* t2    = t1;
    float* out   = (float*)d_out;            // N*64 (+2 tail zeros)

    const int BLK = 256;

    // --- degrees & symmetric norm ------------------------------------------
    fill_ones<<<gridFor(2 * N, BLK), BLK, 0, stream>>>(dinv1, 2 * N);
    edge_degree<<<gridFor(E, BLK), BLK, 0, stream>>>(dst, ew, dinv1, dinv2, E);
    rsqrt_inplace<<<gridFor(2 * N, BLK), BLK, 0, stream>>>(dinv1, 2 * N);

    // --- layer 1: t1 = x@W1 (WMMA), h seeded with self-loop, scatter, bias+relu
    const long long rowTiles = N / 16;       // 6250
    gemm_wmma_f32<8><<<gridFor(rowTiles, 4), 128, 0, stream>>>(x, W1, dinv1, t1, h, (int)N);
    scatter_edges<128><<<gridFor(E * 32, BLK), BLK, 0, stream>>>(t1, src, dst, ew, dinv1, h, E);
    bias_relu<128><<<gridFor(N * 128, BLK), BLK, 0, stream>>>(h, b1, N * 128);

    // --- layer 2: t2 = h@W2 (WMMA), out seeded with self-loop, scatter, bias
    gemm_wmma_f32<4><<<gridFor(rowTiles, 4), 128, 0, stream>>>(h, W2, dinv2, t2, out, (int)N);
    scatter_edges<64><<<gridFor(E * 16, BLK), BLK, 0, stream>>>(t2, src, dst, nullptr, dinv2, out, E);
    bias_tail<64><<<gridFor(N * 64, BLK), BLK, 0, stream>>>(out, b2, N * 64);
}